// Struct2vec_38895223832875
// MI455X (gfx1250) — compile-verified
//
#include <hip/hip_runtime.h>
#include <cstddef>

// ---------------------------------------------------------------------------
// Struct2vec message passing, 1024 nodes, M=128, T=4, on gfx1250 (MI455X).
// Dominant work: 12x (1024x1024)@(1024x128) f32 GEMMs -> V_WMMA_F32_16X16X4_F32.
// Algebraic simplification: edge-MLP is positively homogeneous, so
// G_scaled == G_raw / 1000 -> compute G once, fold 1/1000 into softmax temp.
// B operand is staged in LDS as k-pair-interleaved float2 so each WMMA B
// fetch is one aligned ds_load_b64 (no VGPR repacking moves).
// ---------------------------------------------------------------------------

typedef __attribute__((ext_vector_type(2))) float v2f;
typedef __attribute__((ext_vector_type(4))) float v4f;
typedef __attribute__((ext_vector_type(8))) float v8f;

#define NN   1024
#define MM   128
// pair-row stride in float2 units; 2*144 mod 64(banks) == 32 -> the two lane
// halves (pair p vs p+1) hit disjoint bank sets: conflict-free b64 reads.
#define PSTRIDE 144

// ---------------------------------------------------------------------------
// Stage 1: edge scores.  Gt[v][u] = W1 . relu(D[u,v]*W2[:,0] + dd[u]*W2[:,1]
//                                            + dd[v]*W2[:,2])
// (stored transposed so the softmax over sources u reads contiguous rows)
// ---------------------------------------------------------------------------
__global__ void edge_kernel(const float* __restrict__ D,
                            const float* __restrict__ dd,
                            const float* __restrict__ W1,
                            const float* __restrict__ W2,
                            float* __restrict__ Gt) {
  __shared__ float w0[128], w2c[128], au[128], w1s[128];
  const int u = blockIdx.y;
  const int v = blockIdx.x * 256 + threadIdx.x;
  if (threadIdx.x < 128) {
    const int k = threadIdx.x;
    w0[k]  = W2[k * 3 + 0];
    w2c[k] = W2[k * 3 + 2];
    au[k]  = dd[u] * W2[k * 3 + 1];
    w1s[k] = W1[k];
  }
  __syncthreads();
  const float dv  = dd[v];
  const float Duv = D[(size_t)u * NN + v];
  float acc = 0.f;
#pragma unroll 4
  for (int k = 0; k < 128; ++k) {
    const float h = fmaf(Duv, w0[k], fmaf(dv, w2c[k], au[k]));
    acc = fmaf(fmaxf(h, 0.f), w1s[k], acc);
  }
  Gt[(size_t)v * NN + u] = acc;  // strided store, L2-resident
}

// ---------------------------------------------------------------------------
// Stage 2: per-column softmax (two temperatures), diag masked, + weighted
// distances.  P1 overwrites Gt in place (all reads precede the first barrier).
// ---------------------------------------------------------------------------
__global__ void softmax_kernel(const float* __restrict__ D,
                               float* __restrict__ Gt,   // in: G, out: P1^T
                               float* __restrict__ PT2,
                               float* __restrict__ wd1,
                               float* __restrict__ wd2) {
  __shared__ float r1[256], r2[256];
  const int v = blockIdx.x;
  const int t = threadIdx.x;

  float g[4];
  float m = -1e30f;
#pragma unroll
  for (int i = 0; i < 4; ++i) {
    const int u = t + i * 256;
    float x = Gt[(size_t)v * NN + u];
    if (u == v) x = -1e30f;
    g[i] = x;
    m = fmaxf(m, x);
  }
  r1[t] = m; __syncthreads();
  for (int s = 128; s > 0; s >>= 1) {
    if (t < s) r1[t] = fmaxf(r1[t], r1[t + s]);
    __syncthreads();
  }
  m = r1[0]; __syncthreads();

  const float inv1 = 1.f / (10.f * 1000.f);   // G_scaled/tau == G/(1000*tau)
  const float inv2 = 1.f / 10.f;
  float e1[4], e2[4], s1 = 0.f, s2 = 0.f;
#pragma unroll
  for (int i = 0; i < 4; ++i) {
    const float d = g[i] - m;
    e1[i] = expf(d * inv1);
    e2[i] = expf(d * inv2);
    s1 += e1[i]; s2 += e2[i];
  }
  r1[t] = s1; r2[t] = s2; __syncthreads();
  for (int s = 128; s > 0; s >>= 1) {
    if (t < s) { r1[t] += r1[t + s]; r2[t] += r2[t + s]; }
    __syncthreads();
  }
  const float is1 = 1.f / r1[0];
  const float is2 = 1.f / r2[0];
  __syncthreads();

  float w1p = 0.f, w2p = 0.f;
#pragma unroll
  for (int i = 0; i < 4; ++i) {
    const int u = t + i * 256;
    const float p1 = e1[i] * is1;
    const float p2 = e2[i] * is2;
    Gt[(size_t)v * NN + u]  = p1;
    PT2[(size_t)v * NN + u] = p2;
    const float Duv = D[(size_t)u * NN + v];
    w1p = fmaf(p1, Duv, w1p);
    w2p = fmaf(p2, Duv, w2p);
  }
  r1[t] = w1p; r2[t] = w2p; __syncthreads();
  for (int s = 128; s > 0; s >>= 1) {
    if (t < s) { r1[t] += r1[t + s]; r2[t] += r2[t + s]; }
    __syncthreads();
  }
  if (t == 0) { wd1[v] = r1[0]; wd2[v] = r2[0]; }
}

// ---------------------------------------------------------------------------
// r[v] = wd[v] * relu(mu[v] . Wg).  One wave32 per row.
// ---------------------------------------------------------------------------
__global__ void rowdot_kernel(const float* __restrict__ mu,
                              const float* __restrict__ Wg,
                              const float* __restrict__ wd,
                              float* __restrict__ r) {
  const int wave = threadIdx.x >> 5;
  const int lane = threadIdx.x & 31;
  const int row  = blockIdx.x * 8 + wave;
  const float* mp = mu + (size_t)row * MM + lane * 4;
  const float* wp = Wg + lane * 4;
  float p = mp[0] * wp[0] + mp[1] * wp[1] + mp[2] * wp[2] + mp[3] * wp[3];
  for (int off = 16; off > 0; off >>= 1) p += __shfl_xor(p, off, 32);
  if (lane == 0) r[row] = wd[row] * fmaxf(p, 0.f);
}

// ---------------------------------------------------------------------------
// Unified WMMA GEMM: out(1024 x 128) = act( A(1024 x K) @ Bt + rank-1 terms ).
//   btrans==0: B is row-major KxN (N=128), ldb = row stride       (Pt @ mu)
//   btrans==1: B is a weight matrix, Bt[k][n] = B[n*ldb + koff+k] (Y @ W^T)
// Epilogue fuses: + rvec[row]*wc0[col*ldwc0]  (wd*s * W3[:,0])
//                 + dvec[row]*w4c[col]        (dist * W4)
//                 + bias[row*128+col], optional relu.
// 4 waves/block; each wave owns a 16-row x 128-col strip (8 f32 WMMA tiles).
// A layout per lane: row = lane&15, k-pair = base + (lane<16 ? 0 : 2).
// B layout per lane: VGPR pair = {B[kb][n], B[kb+1][n]}, kb = base+(hi?2:0),
// staged in LDS as interleaved float2 -> single ds_load_b64 per operand.
// ---------------------------------------------------------------------------
__global__ void __launch_bounds__(128)
gemm_fused(const float* __restrict__ A, int lda,
           const float* __restrict__ B, int ldb, int koff, int btrans, int K,
           const float* __restrict__ rvec, const float* __restrict__ wc0, int ldwc0,
           const float* __restrict__ dvec, const float* __restrict__ w4c,
           const float* __restrict__ bias, int do_relu,
           float* __restrict__ out, int ldo) {
  __shared__ v2f Btp[32][PSTRIDE];   // [k/2][col] = {B[k][col], B[k+1][col]}
  const int t    = threadIdx.x;
  const int lane = t & 31;
  const int wave = t >> 5;
  const int n    = lane & 15;
  const int hi   = lane >> 4;                  // lane half selects K pair
  const int rowTile = blockIdx.x * 64 + wave * 16;
  const int rowA    = rowTile + n;

  const v8f zero = {0.f, 0.f, 0.f, 0.f, 0.f, 0.f, 0.f, 0.f};
  v8f acc[8];
#pragma unroll
  for (int j = 0; j < 8; ++j) acc[j] = zero;

  for (int kc = 0; kc < K; kc += 64) {
    __syncthreads();
    if (btrans) {
      // Bt[k][n] = B[n*ldb + koff + kc + k]; thread t == column n.
      const float* src = B + (size_t)t * ldb + koff + kc;
#pragma unroll 8
      for (int k2 = 0; k2 < 32; ++k2) {
        v2f w = {src[2 * k2], src[2 * k2 + 1]};
        Btp[k2][t] = w;
      }
    } else {
      // Row-major K x 128 chunk: thread owns one k-pair x 32-col slab.
      const int rp = t & 31;
      const int ch = (t >> 5) * 32;
      const float* s0 = B + (size_t)(kc + 2 * rp) * ldb + ch;
      const float* s1 = s0 + ldb;
#pragma unroll
      for (int i = 0; i < 32; i += 4) {
        const v4f x0 = *(const v4f*)(s0 + i);
        const v4f x1 = *(const v4f*)(s1 + i);
        v2f p0 = {x0.x, x1.x}, p1 = {x0.y, x1.y};
        v2f p2 = {x0.z, x1.z}, p3 = {x0.w, x1.w};
        Btp[rp][ch + i + 0] = p0;
        Btp[rp][ch + i + 1] = p1;
        Btp[rp][ch + i + 2] = p2;
        Btp[rp][ch + i + 3] = p3;
      }
    }
    __syncthreads();

#pragma unroll 4
    for (int kk = 0; kk < 64; kk += 4) {
      const int p = (kk >> 1) + hi;
      const v2f a = *(const v2f*)(A + (size_t)rowA * lda + kc + kk + hi * 2);
#pragma unroll
      for (int j = 0; j < 8; ++j) {
        const v2f b = Btp[p][j * 16 + n];   // one aligned ds_load_b64
        acc[j] = __builtin_amdgcn_wmma_f32_16x16x4_f32(
            /*neg_a=*/false, a, /*neg_b=*/false, b,
            /*c_mod=*/(short)0, acc[j], /*reuse_a=*/false, /*reuse_b=*/false);
      }
    }
  }

  // Epilogue.  C layout: VGPR i, lane half hi -> row = rowTile + i + hi*8.
#pragma unroll
  for (int j = 0; j < 8; ++j) {
    const int col = j * 16 + n;
    const float cw  = rvec ? wc0[(size_t)col * ldwc0] : 0.f;
    const float w4v = dvec ? w4c[col] : 0.f;
#pragma unroll
    for (int i = 0; i < 8; ++i) {
      const int row = rowTile + i + hi * 8;
      float v = acc[j][i];
      if (rvec) v = fmaf(rvec[row], cw, v);
      if (dvec) v = fmaf(dvec[row], w4v, v);
      if (bias) v += bias[(size_t)row * MM + col];
      if (do_relu) v = fmaxf(v, 0.f);
      out[(size_t)row * ldo + col] = v;
    }
  }
}

// ---------------------------------------------------------------------------
// out = relu(W7 . sum_v mu[v])
// ---------------------------------------------------------------------------
__global__ void final_kernel(const float* __restrict__ mu,
                             const float* __restrict__ W7,
                             float* __restrict__ out) {
  __shared__ float red[128];
  const int j = threadIdx.x;
  float s = 0.f;
  for (int v = 0; v < NN; ++v) s += mu[(size_t)v * MM + j];
  red[j] = s * W7[j];
  __syncthreads();
  for (int st = 64; st > 0; st >>= 1) {
    if (j < st) red[j] += red[j + st];
    __syncthreads();
  }
  if (j == 0) out[0] = fmaxf(red[0], 0.f);
}

// ---------------------------------------------------------------------------
extern "C" void kernel_launch(void* const* d_in, const int* in_sizes, int n_in,
                              void* d_out, int out_size, void* d_ws, size_t ws_size,
                              hipStream_t stream) {
  (void)in_sizes; (void)n_in; (void)out_size; (void)ws_size;

  const float* D    = (const float*)d_in[0];
  const float* dd   = (const float*)d_in[1];
  const float* dr   = (const float*)d_in[2];
  const float* ddep = (const float*)d_in[3];
  const float* mu0A1 = (const float*)d_in[4];
  const float* mu0A2 = (const float*)d_in[5];
  const float* mu0B  = (const float*)d_in[6];
  const float* W1   = (const float*)d_in[7];
  const float* W2   = (const float*)d_in[8];
  const float* W3A1 = (const float*)d_in[9];
  const float* W3A2 = (const float*)d_in[10];
  const float* W4A1 = (const float*)d_in[11];
  const float* W4A2 = (const float*)d_in[12];
  const float* W3B  = (const float*)d_in[13];
  const float* W4B  = (const float*)d_in[14];
  const float* W5   = (const float*)d_in[15];
  const float* W6   = (const float*)d_in[16];
  const float* W7   = (const float*)d_in[17];

  // Workspace layout (~11.6 MB of f32)
  float* ws = (float*)d_ws;
  size_t o = 0;
  float* PT1  = ws + o; o += (size_t)NN * NN;   // G, then P1^T in place
  float* PT2  = ws + o; o += (size_t)NN * NN;
  float* wd1  = ws + o; o += NN;
  float* wd2  = ws + o; o += NN;
  float* rv   = ws + o; o += NN;
  float* mA   = ws + o; o += (size_t)NN * MM;   // mu ping
  float* mB   = ws + o; o += (size_t)NN * MM;   // mu pong
  float* Y    = ws + o; o += (size_t)NN * MM;   // P^T @ mu
  float* AB   = ws + o; o += (size_t)NN * 2 * MM; // [A1 | A2]
  float* bias = ws + o; o += (size_t)NN * MM;

  // 1) edge scores (single pass; scaled variant = /1000 via homogeneity)
  edge_kernel<<<dim3(4, NN), 256, 0, stream>>>(D, dd, W1, W2, PT1);
  // 2) softmaxes + weighted distances
  softmax_kernel<<<NN, 256, 0, stream>>>(D, PT1, PT2, wd1, wd2);

  // 3) layers A1 / A2 (T=4 each), final iteration writes into AB (ld 256)
  auto layerA = [&](const float* PT, const float* wd, const float* mu0,
                    const float* Wg, const float* W3, const float* W4,
                    const float* dist, float* outAB) {
    const float* cur = mu0;
    for (int it = 0; it < 4; ++it) {
      gemm_fused<<<16, 128, 0, stream>>>(PT, NN, cur, MM, 0, 0, NN,
          nullptr, nullptr, 0, nullptr, nullptr, nullptr, 0, Y, MM);
      rowdot_kernel<<<128, 256, 0, stream>>>(cur, Wg, wd, rv);
      float* dst = (it == 3) ? outAB : ((it & 1) ? mB : mA);
      const int ldo = (it == 3) ? 2 * MM : MM;
      // mu' = relu(Y @ W3[:,1:]^T + (wd*s)*W3[:,0] + dist*W4)
      gemm_fused<<<16, 128, 0, stream>>>(Y, MM, W3, MM + 1, 1, 1, MM,
          rv, W3, MM + 1, dist, W4, nullptr, 1, dst, ldo);
      cur = dst;
    }
  };
  layerA(PT1, wd1, mu0A1, W5, W3A1, W4A1, dr,   AB);
  layerA(PT2, wd2, mu0A2, W6, W3A2, W4A2, ddep, AB + MM);

  // 4) bias = [A1|A2] @ W4_B^T   (no relu)
  gemm_fused<<<16, 128, 0, stream>>>(AB, 2 * MM, W4B, 2 * MM, 0, 1, 2 * MM,
      nullptr, nullptr, 0, nullptr, nullptr, nullptr, 0, bias, MM);

  // 5) layer B: mu = relu((P1^T @ mu) @ W3_B^T + bias), T=4
  const float* cur = mu0B;
  for (int it = 0; it < 4; ++it) {
    gemm_fused<<<16, 128, 0, stream>>>(PT1, NN, cur, MM, 0, 0, NN,
        nullptr, nullptr, 0, nullptr, nullptr, nullptr, 0, Y, MM);
    float* dst = (it & 1) ? mB : mA;
    gemm_fused<<<16, 128, 0, stream>>>(Y, MM, W3B, MM, 0, 1, MM,
        nullptr, nullptr, 0, nullptr, nullptr, bias, 1, dst, MM);
    cur = dst;
  }

  // 6) out = relu(W7 . sum_v mu[v])
  final_kernel<<<1, 128, 0, stream>>>(cur, W7, (float*)d_out);
}